// H1Kinematics_82291573391527
// MI455X (gfx1250) — compile-verified
//
#include <hip/hip_runtime.h>

// H1 arm forward kinematics, closed-form reduction.
//
// out = offset - FOREARM*sin(e)*col0(R_sh) - (UPPER_ARM + FOREARM*cos(e))*col2(R_sh)
//   col0 = ( cp*cy + sp*sr*sy,  cr*sy,  cp*sr*sy - sp*cy )
//   col2 = ( sp*cr,            -sr,     cp*cr           )
//
// Memory-bound streaming kernel: 56 MB total traffic, fits in 192 MB L2 across
// graph replays. Native v_sin_f32/v_cos_f32 keep TRANS cost below memory time.

static constexpr float TORSO_H    = 0.42999f;
static constexpr float SHOULDER_W = 0.3107f;
static constexpr float UPPER_ARM  = 0.19886f;
static constexpr float FOREARM    = 0.3f;

__global__ __launch_bounds__(256)
void h1_kinematics_kernel(const float4* __restrict__ qp,
                          const float4* __restrict__ qr,
                          const float4* __restrict__ qy,
                          const float4* __restrict__ qe,
                          float4* __restrict__ out,
                          int n4)
{
    const int t = blockIdx.x * blockDim.x + threadIdx.x;
    if (t >= n4) return;

    // 4 elements per thread: one coalesced b128 load per input array.
    const float4 p = qp[t];
    const float4 r = qr[t];
    const float4 y = qy[t];
    const float4 e = qe[t];

    float o[12];

#pragma unroll
    for (int i = 0; i < 4; ++i) {
        const float pi = (&p.x)[i];
        const float ri = (&r.x)[i];
        const float yi = (&y.x)[i];
        const float ei = (&e.x)[i];

        // Native transcendentals -> v_sin_f32 / v_cos_f32 (inputs ~N(0,1),
        // well inside the accurate range of the HW units).
        const float sp = __sinf(pi), cp = __cosf(pi);
        const float sr = __sinf(ri), cr = __cosf(ri);
        const float sy = __sinf(yi), cy = __cosf(yi);
        const float se = __sinf(ei), ce = __cosf(ei);

        const float k = fmaf(FOREARM, ce, UPPER_ARM);  // UPPER_ARM + FOREARM*ce
        const float a = FOREARM * se;

        const float srsy = sr * sy;
        const float c0x  = fmaf(sp, srsy, cp * cy);
        const float c0y  = cr * sy;
        const float c0z  = fmaf(cp, srsy, -(sp * cy));
        const float c2x  = sp * cr;
        const float c2y  = -sr;
        const float c2z  = cp * cr;

        o[3 * i + 0] =              -fmaf(a, c0x, k * c2x);
        o[3 * i + 1] = SHOULDER_W - fmaf(a, c0y, k * c2y);
        o[3 * i + 2] = TORSO_H    - fmaf(a, c0z, k * c2z);
    }

    // 12 contiguous output floats per thread -> three 16B-aligned b128 stores;
    // the wave's stores tile a contiguous 1536B region (fully coalesced).
    out[3 * t + 0] = make_float4(o[0], o[1], o[2],  o[3]);
    out[3 * t + 1] = make_float4(o[4], o[5], o[6],  o[7]);
    out[3 * t + 2] = make_float4(o[8], o[9], o[10], o[11]);
}

extern "C" void kernel_launch(void* const* d_in, const int* in_sizes, int n_in,
                              void* d_out, int out_size, void* d_ws, size_t ws_size,
                              hipStream_t stream) {
    const int n  = in_sizes[0];   // B = 2097152 (divisible by 4)
    const int n4 = n >> 2;

    const float4* qp = (const float4*)d_in[0];
    const float4* qr = (const float4*)d_in[1];
    const float4* qy = (const float4*)d_in[2];
    const float4* qe = (const float4*)d_in[3];
    float4* out = (float4*)d_out;

    const int block = 256;                         // 8 wave32s per block
    const int grid  = (n4 + block - 1) / block;    // 2048 blocks

    h1_kinematics_kernel<<<grid, block, 0, stream>>>(qp, qr, qy, qe, out, n4);
}